// BaseTransducer_41970420417852
// MI455X (gfx1250) — compile-verified
//
#include <hip/hip_runtime.h>
#include <hip/hip_bf16.h>

typedef __attribute__((ext_vector_type(2))) float v2f;
typedef __attribute__((ext_vector_type(4))) float v4f;
typedef __attribute__((ext_vector_type(8))) float v8f;

// -----------------------------------------------------------------------------
// Kernel 1: S[m, v] = A[m, :] . W[v, :]  (+ bias[v] for the text rows)
// A is the row-concatenation of speech [B*T, D] and text [B*U, D].
// One wave computes one 16x16 output tile via V_WMMA_F32_16X16X4_F32,
// looping K over D in steps of 4.
//
// f32 WMMA operand layout (ISA 7.12.2):
//   A (16x4): lane<16 holds row M=lane, VGPR0=K0,VGPR1=K1; lane>=16 holds K2,K3
//     -> contiguous float2 at A[row, 4k + 2*(lane>>4)]
//   B (4x16): mirrored with N=lane&15 -> contiguous float2 at W[col, 4k + 2*half]
//   C/D: VGPR r -> M = r + 8*(lane>>4), N = lane&15
// -----------------------------------------------------------------------------
__global__ void __launch_bounds__(256)
rnnt_proj_wmma(const float* __restrict__ speech,
               const float* __restrict__ text,
               const float* __restrict__ W,
               const float* __restrict__ bias,
               float* __restrict__ S,      // [Mtot, V]
               int Mspeech, int Mtot, int D, int V)
{
    const int waveId = (blockIdx.x * blockDim.x + threadIdx.x) >> 5;
    const int lane   = threadIdx.x & 31;
    const int tilesN = V >> 4;
    const int tileM  = waveId / tilesN;
    const int tileN  = waveId - tileM * tilesN;
    const int row0   = tileM << 4;
    const int col0   = tileN << 4;
    if (row0 >= Mtot) return;

    const int idx  = lane & 15;
    const int half = lane >> 4;

    // Select source region (tiles never straddle: Mspeech is a multiple of 16).
    const float* Abase = (row0 < Mspeech)
        ? (speech + (size_t)row0 * D)
        : (text + (size_t)(row0 - Mspeech) * D);

    const float* aptr = Abase + (size_t)idx * D + 2 * half;
    const float* bptr = W + (size_t)(col0 + idx) * D + 2 * half;

    v8f acc = {};
#pragma unroll 4
    for (int k = 0; k < D; k += 4) {
        v2f afrag = *(const v2f*)(aptr + k);
        v2f bfrag = *(const v2f*)(bptr + k);
        acc = __builtin_amdgcn_wmma_f32_16x16x4_f32(
            /*neg_a=*/false, afrag, /*neg_b=*/false, bfrag,
            /*c_mod=*/(short)0, acc, /*reuse_a=*/false, /*reuse_b=*/false);
    }

    // Fold bias into the text half so the broadcast kernel is a pure add.
    const float bb = (row0 >= Mspeech) ? bias[col0 + idx] : 0.0f;

    float* out = S + (size_t)(row0 + 8 * half) * V + col0 + idx;
#pragma unroll
    for (int r = 0; r < 8; ++r) {
        out[(size_t)r * V] = acc[r] + bb;
    }
}

// -----------------------------------------------------------------------------
// Kernel 2: out[b,t,u,v] = S[b*T+t, v] + Xtb[b*U+u, v]
// float4 per thread; S/Xtb (5 MB) stay L2-resident, output (512 MB) uses
// non-temporal stores to avoid flushing L2.
// -----------------------------------------------------------------------------
__global__ void __launch_bounds__(256)
rnnt_bcast_add(const v4f* __restrict__ S,    // [B*T, V/4]
               const v4f* __restrict__ Xtb,  // [B*U, V/4]
               v4f* __restrict__ out,        // [B*T*U, V/4]
               int T, int U, int V4, long long total4)
{
    long long p = (long long)blockIdx.x * blockDim.x + threadIdx.x;
    if (p >= total4) return;

    const int v       = (int)(p % V4);
    long long r       = p / V4;           // = ((b*T + t)*U + u)
    const int u       = (int)(r % U);
    long long bt      = r / U;            // = b*T + t
    const int t       = (int)(bt % T);
    const int b       = (int)(bt / T);

    v4f s = S[(size_t)bt * V4 + v];
    v4f x = Xtb[((size_t)b * U + u) * V4 + v];
    v4f o = s + x;
    __builtin_nontemporal_store(o, out + p);
}

extern "C" void kernel_launch(void* const* d_in, const int* in_sizes, int n_in,
                              void* d_out, int out_size, void* d_ws, size_t ws_size,
                              hipStream_t stream) {
    const float* speech = (const float*)d_in[0];
    const float* text   = (const float*)d_in[1];
    const float* W      = (const float*)d_in[2];
    const float* bias   = (const float*)d_in[3];
    const int*   slen   = (const int*)d_in[4];
    const int*   tlen   = (const int*)d_in[5];

    const int B = in_sizes[4];              // len(speech_len)
    const int V = in_sizes[3];              // len(b)
    const int D = in_sizes[2] / V;          // W is [V, D]
    const int T = in_sizes[0] / (B * D);    // speech [B,T,D]
    const int U = in_sizes[1] / (B * D);    // text   [B,U,D]

    const int Mspeech = B * T;
    const int Mtext   = B * U;
    const int Mtot    = Mspeech + Mtext;

    float* S   = (float*)d_ws;                          // [Mtot, V] intermediates
    float* Xtb = S + (size_t)Mspeech * V;               // text half (bias folded)

    // --- GEMM: one wave per 16x16 tile, 8 waves per 256-thread block ---
    const long long tiles  = (long long)(Mtot / 16) * (V / 16);
    const int gemmBlocks   = (int)((tiles + 7) / 8);
    rnnt_proj_wmma<<<gemmBlocks, 256, 0, stream>>>(
        speech, text, W, bias, S, Mspeech, Mtot, D, V);

    // --- Broadcast add (output-store bound: ~512 MB @ 23.3 TB/s) ---
    const int V4 = V / 4;
    const long long total4 = (long long)B * T * U * V4;
    const int bcBlocks = (int)((total4 + 255) / 256);
    rnnt_bcast_add<<<bcBlocks, 256, 0, stream>>>(
        (const v4f*)S, (const v4f*)Xtb, (v4f*)d_out, T, U, V4, total4);

    // --- Tuple tail: speech_len / text_len appended after logits (4B elems) ---
    float* outf = (float*)d_out;
    size_t logitsN = (size_t)B * T * U * V;
    hipMemcpyAsync(outf + logitsN,     slen, (size_t)B * sizeof(int),
                   hipMemcpyDeviceToDevice, stream);
    hipMemcpyAsync(outf + logitsN + B, tlen, (size_t)B * sizeof(int),
                   hipMemcpyDeviceToDevice, stream);
}